// VoxelFeatureEncoder_84765474554591
// MI455X (gfx1250) — compile-verified
//
#include <hip/hip_runtime.h>
#include <hip/hip_bf16.h>

typedef __attribute__((ext_vector_type(2))) float    v2f;
typedef __attribute__((ext_vector_type(8))) float    v8f;
typedef __attribute__((ext_vector_type(4))) unsigned v4u;
typedef __attribute__((ext_vector_type(4))) int      v4i;
typedef __attribute__((ext_vector_type(8))) int      v8i;

#define VN        262144
#define GRID_NX   500
#define GRID_NY   500
#define GRID_NZ   30
#define GRID_B    2
#define SENTINEL  (GRID_B * GRID_NX * GRID_NY * GRID_NZ)   /* 15,000,000 */
#define SCAN_TILE 4096
#define NSCANBLK  ((SENTINEL + SCAN_TILE - 1) / SCAN_TILE) /* 3663 */

// Dynamic-LDS float offsets for the fused MLP kernel (~213 KB of 320 KB/WGP)
#define SM_W1   0                    //   256 f  (4x64)
#define SM_W2   (SM_W1 + 256)        //  8192 f  (64x128)
#define SM_W3   (SM_W2 + 8192)       // 32768 f  (128x256)
#define SM_A0   (SM_W3 + 32768)      //   512 f  (64 rows, pitch 8)
#define SM_A1   (SM_A0 + 512)        //  4352 f  (64 rows, pitch 68)
#define SM_A2   (SM_A1 + 4352)       //  8448 f  (64 rows, pitch 132)
#define SM_TOT  (SM_A2 + 8448)       // 54528 floats = 218,112 bytes

// ---------------------------------------------------------------------------
// K1: zero the dense occupancy/scan array + init per-voxel-slot buffers
// ---------------------------------------------------------------------------
__global__ void k_init(unsigned* __restrict__ scan_arr,
                       unsigned* __restrict__ bsum,
                       unsigned* __restrict__ uniq,
                       float* __restrict__ sums,
                       float* __restrict__ cnt) {
    int i = blockIdx.x * blockDim.x + threadIdx.x;
    if (i < SENTINEL) scan_arr[i] = 0u;
    if (i < SCAN_TILE) bsum[i] = 0u;
    if (i < VN) {
        uniq[i] = (unsigned)SENTINEL;
        cnt[i] = 0.0f;
        sums[i * 4 + 0] = 0.0f;
        sums[i * 4 + 1] = 0.0f;
        sums[i * 4 + 2] = 0.0f;
        sums[i * 4 + 3] = 0.0f;
    }
}

// ---------------------------------------------------------------------------
// K2: voxelize each point -> flat key; mark occupancy
// ---------------------------------------------------------------------------
__global__ void k_voxelize(const float* __restrict__ points,
                           const int* __restrict__ coors,
                           unsigned* __restrict__ keys,
                           unsigned* __restrict__ scan_arr) {
    int i = blockIdx.x * blockDim.x + threadIdx.x;
    if (i >= VN) return;
    float px = points[i * 4 + 0];
    float py = points[i * 4 + 1];
    float pz = points[i * 4 + 2];
    int vx = (int)floorf((px - (-50.0f)) / 0.2f);
    int vy = (int)floorf((py - (-50.0f)) / 0.2f);
    int vz = (int)floorf((pz - (-3.0f)) / 0.2f);
    vx = vx < 0 ? 0 : (vx > GRID_NX - 1 ? GRID_NX - 1 : vx);
    vy = vy < 0 ? 0 : (vy > GRID_NY - 1 ? GRID_NY - 1 : vy);
    vz = vz < 0 ? 0 : (vz > GRID_NZ - 1 ? GRID_NZ - 1 : vz);
    int b = coors[i * 3 + 0];
    unsigned key = (unsigned)(((b * GRID_NX + vx) * GRID_NY + vy) * GRID_NZ + vz);
    keys[i] = key;
    scan_arr[key] = 1u;   // benign race: all writers store 1
}

// ---------------------------------------------------------------------------
// K3: in-place exclusive scan of occupancy, per 4096-element block
// ---------------------------------------------------------------------------
__global__ __launch_bounds__(256)
void k_scan_blocks(unsigned* __restrict__ data, unsigned* __restrict__ bsum) {
    __shared__ unsigned tsum[256];
    const int tid = threadIdx.x;
    const int base = blockIdx.x * SCAN_TILE + tid * 16;
    unsigned v[16];
    unsigned s = 0;
    #pragma unroll
    for (int j = 0; j < 16; ++j) {
        int idx = base + j;
        unsigned x = (idx < SENTINEL) ? data[idx] : 0u;
        v[j] = s;          // exclusive within thread chunk
        s += x;
    }
    tsum[tid] = s;
    __syncthreads();
    for (int off = 1; off < 256; off <<= 1) {
        unsigned t = (tid >= off) ? tsum[tid - off] : 0u;
        __syncthreads();
        tsum[tid] += t;
        __syncthreads();
    }
    unsigned excl = (tid == 0) ? 0u : tsum[tid - 1];
    if (tid == 255) bsum[blockIdx.x] = tsum[255];
    #pragma unroll
    for (int j = 0; j < 16; ++j) {
        int idx = base + j;
        if (idx < SENTINEL) data[idx] = v[j] + excl;
    }
}

// K4: serial exclusive scan of the 3663 block sums (trivial cost)
__global__ void k_scan_bsums(unsigned* __restrict__ bsum, int n) {
    if (blockIdx.x == 0 && threadIdx.x == 0) {
        unsigned acc = 0;
        for (int i = 0; i < n; ++i) {
            unsigned t = bsum[i];
            bsum[i] = acc;
            acc += t;
        }
    }
}

// ---------------------------------------------------------------------------
// K5: rank lookup (= sorted-unique index), record uniq value, segment sums
// ---------------------------------------------------------------------------
__global__ void k_scatter(const float* __restrict__ points,
                          const unsigned* __restrict__ keys,
                          const unsigned* __restrict__ scan_arr,
                          const unsigned* __restrict__ bsum,
                          unsigned* __restrict__ uniq,
                          float* __restrict__ sums,
                          float* __restrict__ cnt) {
    int i = blockIdx.x * blockDim.x + threadIdx.x;
    if (i >= VN) return;
    unsigned k = keys[i];
    unsigned rank = scan_arr[k] + bsum[k >> 12];   // 4096 = 2^12
    uniq[rank] = k;                                 // idempotent
    atomicAdd(&sums[rank * 4 + 0], points[i * 4 + 0]);
    atomicAdd(&sums[rank * 4 + 1], points[i * 4 + 1]);
    atomicAdd(&sums[rank * 4 + 2], points[i * 4 + 2]);
    atomicAdd(&sums[rank * 4 + 3], points[i * 4 + 3]);
    atomicAdd(&cnt[rank], 1.0f);
}

// ---------------------------------------------------------------------------
// TDM: 1-D DMA of `nelem` fp32 from global into LDS (CDNA5 tensor_load_to_lds)
//   D# group0: count=1, lds_addr, 57-bit global_addr, type=2 ("image")
//   D# group1: data_size=2 (4B), tensor_dim0=nelem, tensor_dim1=1,
//              tile_dim0=nelem (<=32768 fits 16-bit), tile_dim1=1, stride0=nelem
//   Completion tracked on TENSORcnt.
// ---------------------------------------------------------------------------
__device__ __forceinline__ void tdm_load_f32(const float* __restrict__ gsrc,
                                             float* __restrict__ ldst,
                                             unsigned nelem) {
    unsigned lds_off = (unsigned)(size_t)ldst;                  // addr[31:0] = LDS offset
    unsigned long long ga = (unsigned long long)(size_t)gsrc;   // 57-bit global address
    v4u g0 = { 1u,                                              // count=1, user mode
               lds_off,                                         // bits 63:32
               (unsigned)(ga & 0xffffffffull),                  // bits 95:64
               (unsigned)((ga >> 32) & 0x01ffffffull) | 0x80000000u }; // addr[56:32] | type=2
    v8i g1 = { (int)(2u << 16),                                 // data_size=4B; wg_mask=0
               (int)((nelem & 0xffffu) << 16),                  // tensor_dim0[15:0] @ bit48
               (int)(((nelem >> 16) & 0xffffu) | (1u << 16)),   // tensor_dim0[31:16] | tensor_dim1[15:0]
               (int)((nelem & 0xffffu) << 16),                  // tensor_dim1[31:16]=0 | tile_dim0
               1,                                               // tile_dim1=1, tile_dim2=0
               (int)nelem,                                      // tensor_dim0_stride[31:0]
               0, 0 };
    v4i gz = { 0, 0, 0, 0 };
#if defined(__clang_major__) && (__clang_major__ >= 23)
    v8i gz8 = { 0, 0, 0, 0, 0, 0, 0, 0 };
    __builtin_amdgcn_tensor_load_to_lds(g0, g1, gz, gz, gz8, 0);
#else
    __builtin_amdgcn_tensor_load_to_lds(g0, g1, gz, gz, 0);
#endif
}

// ---------------------------------------------------------------------------
// K6: fused 3-layer MLP on WMMA fp32 matrix pipe, weights TDM-staged in LDS.
//   64-row tiles, 4 waves x 16 rows; activations ping-pong in padded LDS.
//   V_WMMA_F32_16X16X4_F32 lane layout (ISA 7.12.2):
//     A 16x4 : lanes 0-15 M=lane hold K={k0,k0+1}; lanes 16-31 K={k0+2,k0+3}
//     B 4x16 : lanes 0-15 N=lane hold K={k0,k0+1}; lanes 16-31 K={k0+2,k0+3}
//     C 16x16: VGPR r -> M = r (+8 for lanes 16-31), N = lane&15
//   A-fragments hoisted to registers (reused across all Nout/16 column tiles).
// ---------------------------------------------------------------------------
template <int Kin, int Nout>
__device__ __forceinline__ void mlp_layer_wmma(
    const float* __restrict__ actPrev, int pitchPrev,
    const float* __restrict__ Wlds,
    const float* __restrict__ g, const float* __restrict__ bv,
    float* __restrict__ actNext, int pitchNext,
    int wave, int lane, float inv_std) {
    const int rbase  = wave * 16;
    const int l15    = lane & 15;
    const int khalf  = (lane >> 4) << 1;             // 0 or 2
    const int rowOff = rbase + ((lane >> 4) << 3);   // +8 for upper lane half

    v2f afrag[Kin / 4];
    #pragma unroll
    for (int kk = 0; kk < Kin / 4; ++kk) {
        const float* ap = actPrev + (rbase + l15) * pitchPrev + kk * 4 + khalf;
        afrag[kk].x = ap[0];
        afrag[kk].y = ap[1];
    }
    for (int ct = 0; ct < Nout; ct += 16) {
        v8f acc = {};
        #pragma unroll
        for (int kk = 0; kk < Kin / 4; ++kk) {
            v2f b;
            b.x = Wlds[(kk * 4 + khalf) * Nout + ct + l15];
            b.y = Wlds[(kk * 4 + khalf + 1) * Nout + ct + l15];
            acc = __builtin_amdgcn_wmma_f32_16x16x4_f32(
                false, afrag[kk], false, b, (short)0, acc, false, false);
        }
        float sc = g[ct + l15] * inv_std;
        float bi = bv[ct + l15];
        #pragma unroll
        for (int r = 0; r < 8; ++r) {
            float v = acc[r] * sc + bi;
            v = v > 0.0f ? v : 0.0f;
            actNext[(rowOff + r) * pitchNext + ct + l15] = v;
        }
    }
}

__global__ __launch_bounds__(128)
void k_mlp_fused(const float* __restrict__ sums, const float* __restrict__ cnt,
                 const float* __restrict__ W1, const float* __restrict__ g1,
                 const float* __restrict__ b1,
                 const float* __restrict__ W2, const float* __restrict__ g2,
                 const float* __restrict__ b2,
                 const float* __restrict__ W3, const float* __restrict__ g3,
                 const float* __restrict__ b3,
                 float* __restrict__ out) {
    extern __shared__ float smem[];
    float* w1s = smem + SM_W1;
    float* w2s = smem + SM_W2;
    float* w3s = smem + SM_W3;
    float* a0  = smem + SM_A0;
    float* a1  = smem + SM_A1;
    float* a2  = smem + SM_A2;

    const int gRow0 = blockIdx.x * 64;
    const int tid = threadIdx.x;

    // Wave 0 kicks off the weight DMAs (TDM ignores EXEC; one issue per block)
    if (tid < 32) {
        tdm_load_f32(W1, w1s, 256);
        tdm_load_f32(W2, w2s, 8192);
        tdm_load_f32(W3, w3s, 32768);
    }

    // Overlap: all threads stage voxel_feats = sums / max(cnt, 1)
    for (int idx = tid; idx < 64 * 4; idx += 128) {
        int r = idx >> 2, c = idx & 3;
        float cn = cnt[gRow0 + r];
        float denom = cn > 1.0f ? cn : 1.0f;
        a0[r * 8 + c] = sums[(gRow0 + r) * 4 + c] / denom;
    }
    if (tid < 32) __builtin_amdgcn_s_wait_tensorcnt(0);
    __syncthreads();   // weights + act0 now visible to all 4 waves

    const int wave = tid >> 5, lane = tid & 31;
    const float inv_std = rsqrtf(1.0f + 1e-5f);

    mlp_layer_wmma<4, 64>(a0, 8, w1s, g1, b1, a1, 68, wave, lane, inv_std);
    __syncthreads();
    mlp_layer_wmma<64, 128>(a1, 68, w2s, g2, b2, a2, 132, wave, lane, inv_std);
    __syncthreads();
    mlp_layer_wmma<128, 256>(a2, 132, w3s, g3, b3,
                             out + (size_t)gRow0 * 256, 256, wave, lane, inv_std);
}

// ---------------------------------------------------------------------------
// K7: decode unique ids -> coords, appended after the N*256 activations
// ---------------------------------------------------------------------------
__global__ void k_write_coors(const unsigned* __restrict__ uniq,
                              float* __restrict__ out_tail) {
    int i = blockIdx.x * blockDim.x + threadIdx.x;
    if (i >= VN) return;
    unsigned u = uniq[i];
    unsigned z = u % GRID_NZ;
    unsigned y = (u / GRID_NZ) % GRID_NY;
    unsigned x = (u / (GRID_NZ * GRID_NY)) % GRID_NX;
    unsigned b = u / (GRID_NZ * GRID_NY * GRID_NX);
    out_tail[i * 4 + 0] = (float)b;   // values < 2^24: lossless in fp32
    out_tail[i * 4 + 1] = (float)x;
    out_tail[i * 4 + 2] = (float)y;
    out_tail[i * 4 + 3] = (float)z;
}

// ---------------------------------------------------------------------------
extern "C" void kernel_launch(void* const* d_in, const int* in_sizes, int n_in,
                              void* d_out, int out_size, void* d_ws, size_t ws_size,
                              hipStream_t stream) {
    const float* points = (const float*)d_in[0];
    const int*   coors  = (const int*)d_in[1];
    const float* W1 = (const float*)d_in[2];
    const float* g1 = (const float*)d_in[3];
    const float* b1 = (const float*)d_in[4];
    const float* W2 = (const float*)d_in[5];
    const float* g2 = (const float*)d_in[6];
    const float* b2 = (const float*)d_in[7];
    const float* W3 = (const float*)d_in[8];
    const float* g3 = (const float*)d_in[9];
    const float* b3 = (const float*)d_in[10];
    float* out = (float*)d_out;

    // workspace layout (~67.4 MB)
    unsigned* scan_arr = (unsigned*)d_ws;          // SENTINEL u32   (60 MB)
    unsigned* bsum     = scan_arr + SENTINEL;      // 4096 u32
    unsigned* keys     = bsum + SCAN_TILE;         // VN u32
    unsigned* uniq     = keys + VN;                // VN u32
    float*    sums     = (float*)(uniq + VN);      // VN*4 f32
    float*    cnt      = sums + (size_t)VN * 4;    // VN f32

    const int T = 256;
    k_init<<<(SENTINEL + T - 1) / T, T, 0, stream>>>(scan_arr, bsum, uniq, sums, cnt);
    k_voxelize<<<(VN + T - 1) / T, T, 0, stream>>>(points, coors, keys, scan_arr);
    k_scan_blocks<<<NSCANBLK, 256, 0, stream>>>(scan_arr, bsum);
    k_scan_bsums<<<1, 1, 0, stream>>>(bsum, NSCANBLK);
    k_scatter<<<(VN + T - 1) / T, T, 0, stream>>>(points, keys, scan_arr, bsum,
                                                  uniq, sums, cnt);
    k_mlp_fused<<<VN / 64, 128, SM_TOT * sizeof(float), stream>>>(
        sums, cnt, W1, g1, b1, W2, g2, b2, W3, g3, b3, out);
    k_write_coors<<<(VN + T - 1) / T, T, 0, stream>>>(uniq, out + (size_t)VN * 256);
}